// ConstrainedAttentionModel_38517266710953
// MI455X (gfx1250) — compile-verified
//
#include <hip/hip_runtime.h>
#include <hip/hip_bf16.h>
#include <math.h>

typedef __attribute__((ext_vector_type(2))) float v2f;
typedef __attribute__((ext_vector_type(8))) float v8f;

#define T_LEN 2048
#define NTHREADS 256
#define NWAVES (NTHREADS / 32)
#define CHUNKS (T_LEN / 16)               // 128 tiles of 16 t-values
#define CHUNKS_PER_WAVE (CHUNKS / NWAVES) // 16

__global__ __launch_bounds__(NTHREADS)
void constrained_attn_kernel(const int* __restrict__ x,
                             const float* __restrict__ C,
                             float* __restrict__ out) {
  __shared__ int   s_x[T_LEN];        // token row (8 KB)
  __shared__ float s_scores[T_LEN];   // scores -> exp(scores) (8 KB)
  __shared__ float s_out[T_LEN];      // output histogram (8 KB)
  __shared__ float s_C[16];
  __shared__ float s_tmp[NTHREADS];

  const int tid  = threadIdx.x;
  const int lane = tid & 31;
  const int wave = tid >> 5;
  const int b    = blockIdx.x;

  // ---- stage inputs into LDS ----
  const int* xrow = x + b * T_LEN;
  for (int t = tid; t < T_LEN; t += NTHREADS) s_x[t] = xrow[t];
  if (tid < 16) s_C[tid] = C[tid];
  __syncthreads();

  // query tokens q_i = x[b, T-1-i]
  int qt[4];
  #pragma unroll
  for (int i = 0; i < 4; ++i) qt[i] = s_x[T_LEN - 1 - i];

  // Per-lane j assignment follows the 32-bit 16x4 A-matrix layout:
  // lanes 0-15 hold K(=j) 0 (v0) and 1 (v1); lanes 16-31 hold j 2 and 3.
  const int jlo = (lane < 16) ? 0 : 2;
  float cA[4], cB[4];
  #pragma unroll
  for (int g = 0; g < 4; ++g) {
    cA[g] = s_C[g * 4 + jlo];
    cB[g] = s_C[g * 4 + jlo + 1];
  }

  const int m = lane & 15; // A-matrix row within the 16-token tile
  v2f bones; bones.x = 1.0f; bones.y = 1.0f; // all-ones B (layout-agnostic)

  // ---- scores via 4 chained V_WMMA_F32_16X16X4_F32 per 16-token tile ----
  for (int cc = wave * CHUNKS_PER_WAVE; cc < (wave + 1) * CHUNKS_PER_WAVE; ++cc) {
    const int t = cc * 16 + m;
    const int vA = (t >= jlo)     ? s_x[t - jlo]     : -1;
    const int vB = (t >= jlo + 1) ? s_x[t - jlo - 1] : -1;

    v8f acc = {};
    #pragma unroll
    for (int g = 0; g < 4; ++g) {
      v2f a;
      a.x = (vA == qt[g]) ? cA[g] : 0.0f; // C folded into A; B is ones
      a.y = (vB == qt[g]) ? cB[g] : 0.0f;
      acc = __builtin_amdgcn_wmma_f32_16x16x4_f32(
          /*neg_a=*/false, a, /*neg_b=*/false, bones,
          /*c_mod=*/(short)0, acc, /*reuse_a=*/false, /*reuse_b=*/false);
    }

    // D layout: VGPR r holds M=r (lanes 0-15, N=lane) and M=8+r (lanes 16-31).
    // All 16 columns are identical; lanes 0 and 16 write rows 0-7 / 8-15.
    if (m == 0) {
      const int base = cc * 16 + ((lane >> 4) << 3);
      #pragma unroll
      for (int r = 0; r < 8; ++r) s_scores[base + r] = acc[r];
    }
  }
  __syncthreads();

  // ---- mask last position, zero the output histogram ----
  if (tid == 0) s_scores[T_LEN - 1] = -1.0e9f;
  for (int v = tid; v < T_LEN; v += NTHREADS) s_out[v] = 0.0f;
  __syncthreads();

  // ---- softmax: block max ----
  float lmax = -INFINITY;
  for (int t = tid; t < T_LEN; t += NTHREADS) lmax = fmaxf(lmax, s_scores[t]);
  s_tmp[tid] = lmax;
  __syncthreads();
  for (int s = NTHREADS / 2; s > 0; s >>= 1) {
    if (tid < s) s_tmp[tid] = fmaxf(s_tmp[tid], s_tmp[tid + s]);
    __syncthreads();
  }
  const float bmax = s_tmp[0];
  __syncthreads();

  // ---- softmax: exp + block sum (exp cached in s_scores) ----
  float lsum = 0.0f;
  for (int t = tid; t < T_LEN; t += NTHREADS) {
    const float e = __expf(s_scores[t] - bmax);
    s_scores[t] = e;
    lsum += e;
  }
  s_tmp[tid] = lsum;
  __syncthreads();
  for (int s = NTHREADS / 2; s > 0; s >>= 1) {
    if (tid < s) s_tmp[tid] += s_tmp[tid + s];
    __syncthreads();
  }
  const float inv = 1.0f / s_tmp[0];
  __syncthreads();

  // ---- scatter attention weights into the vocab histogram (ds_add_f32) ----
  for (int t = tid; t < T_LEN; t += NTHREADS) {
    atomicAdd(&s_out[s_x[t]], s_scores[t] * inv);
  }
  __syncthreads();

  // ---- write result row ----
  float* orow = out + b * T_LEN;
  for (int v = tid; v < T_LEN; v += NTHREADS) orow[v] = s_out[v];
}

extern "C" void kernel_launch(void* const* d_in, const int* in_sizes, int n_in,
                              void* d_out, int out_size, void* d_ws, size_t ws_size,
                              hipStream_t stream) {
  const int*   x   = (const int*)d_in[0];   // (8, 2048) tokens
  const float* C   = (const float*)d_in[1]; // (4, 4) row-major
  float*       out = (float*)d_out;         // (8, 2048)
  constrained_attn_kernel<<<dim3(8), dim3(NTHREADS), 0, stream>>>(x, C, out);
}